// DiScoWeightedLoss_55035710931339
// MI455X (gfx1250) — compile-verified
//
#include <hip/hip_runtime.h>
#include <hip/hip_bf16.h>
#include <math.h>

#define DISCO_LAMBDA 0.1f
#define EPS_W   1e-8f
#define EPS_VAR 1e-10f

typedef float v2f __attribute__((ext_vector_type(2)));
typedef float v8f __attribute__((ext_vector_type(8)));

// Exact wave32 sum via V_WMMA_F32_16X16X4_F32.
// A layout (documented): lane m (0..15) holds A[m][0]=a.x, A[m][1]=a.y;
// lane m+16 holds A[m][2]=a.x, A[m][3]=a.y.  With a.y=0 and B == all-ones
// (layout-agnostic), D[m][n] = x_m + x_{m+16}.  C layout (documented):
// lane n<16 holds rows 0..7 in c[0..7], lanes 16..31 hold rows 8..15.
// Summing c[0..7] gives each half-sum; shfl_xor(16) completes the wave sum.
__device__ __forceinline__ float wave_sum_wmma(float x) {
  v2f a; a[0] = x;    a[1] = 0.0f;
  v2f b; b[0] = 1.0f; b[1] = 1.0f;
  v8f c = {0.f,0.f,0.f,0.f,0.f,0.f,0.f,0.f};
  c = __builtin_amdgcn_wmma_f32_16x16x4_f32(false, a, false, b, (short)0, c,
                                            false, false);
  float s = ((c[0]+c[1]) + (c[2]+c[3])) + ((c[4]+c[5]) + (c[6]+c[7]));
  s += __shfl_xor(s, 16, 32);
  return s;
}

// CDNA5 async global->LDS copy, 16B per lane (ASYNCcnt tracked).
__device__ __forceinline__ void async_g2l_b128(const float* g, const float* l) {
  unsigned lds = (unsigned)(unsigned long long)l; // low 32 bits = LDS offset
  asm volatile("global_load_async_to_lds_b128 %0, %1, off"
               :: "v"(lds), "v"(g) : "memory");
}
__device__ __forceinline__ void wait_asynccnt0() {
  asm volatile("s_wait_asynccnt 0" ::: "memory");
}

// ---------------- Kernel 1: CE + scores + masked weights + partial sums ----
__global__ void k_prep(const float* __restrict__ logits,
                       const int*   __restrict__ target,
                       const float* __restrict__ weight,
                       const float* __restrict__ mass1,
                       const float* __restrict__ mass2,
                       float* __restrict__ scores,
                       float* __restrict__ w1o,
                       float* __restrict__ w2o,
                       float* __restrict__ prep_part) {
  __shared__ float red[8][4];
  const int i = blockIdx.x * 256 + threadIdx.x;
  float l0 = logits[3*i+0], l1 = logits[3*i+1], l2 = logits[3*i+2];
  float mx  = fmaxf(l0, fmaxf(l1, l2));
  float e0  = expf(l0-mx), e1 = expf(l1-mx), e2 = expf(l2-mx);
  float sum = e0 + e1 + e2;
  float lse = mx + logf(sum);
  int   t   = target[i];
  float lt  = (t == 0) ? l0 : ((t == 1) ? l1 : l2);
  float w   = weight[i];
  float ce  = lse - lt;
  float sc  = e0 / sum;                       // softmax class-0 score
  float w1  = (mass1[i] > 0.f) ? w : 0.f;
  float w2  = (mass2[i] > 0.f) ? w : 0.f;
  scores[i] = sc;  w1o[i] = w1;  w2o[i] = w2;

  float v[4] = { ce * w, w, w1, w2 };
  const int wave = threadIdx.x >> 5, lane = threadIdx.x & 31;
  #pragma unroll
  for (int q = 0; q < 4; ++q) {
    float s = wave_sum_wmma(v[q]);
    if (lane == 0) red[wave][q] = s;
  }
  __syncthreads();
  if (threadIdx.x == 0) {
    #pragma unroll
    for (int q = 0; q < 4; ++q) {
      float s = 0.f;
      for (int wv = 0; wv < 8; ++wv) s += red[wv][q];
      prep_part[blockIdx.x * 4 + q] = s;
    }
  }
}

// ---------------- Kernel 2: fused O(N^2) DisCo pass (both terms) -----------
// Per i: m^x = sum_j |x_i-x_j| w_j,  m^y likewise, and the three quadratic
// sums q_xx, q_xy, q_yy — double-centering identity removes all other passes.
__global__ void k_pairs(const float* __restrict__ scores_g,
                        const float* __restrict__ m1_g,
                        const float* __restrict__ m2_g,
                        const float* __restrict__ w1_g,
                        const float* __restrict__ w2_g,
                        const float* __restrict__ prep_part, int nbPrep,
                        int N,
                        float* __restrict__ pair_part) {
  extern __shared__ float smem[];
  float* s_s  = smem;
  float* s_y1 = smem + N;
  float* s_y2 = smem + 2*N;
  float* s_u1 = smem + 3*N;
  float* s_u2 = smem + 4*N;
  const int tid = threadIdx.x;

  // Async-stage all five arrays into LDS (80KB for N=4096; WGP has 320KB).
  for (int idx = tid * 4; idx < N; idx += blockDim.x * 4) {
    async_g2l_b128(scores_g + idx, s_s  + idx);
    async_g2l_b128(m1_g     + idx, s_y1 + idx);
    async_g2l_b128(m2_g     + idx, s_y2 + idx);
    async_g2l_b128(w1_g     + idx, s_u1 + idx);
    async_g2l_b128(w2_g     + idx, s_u2 + idx);
  }
  wait_asynccnt0();
  __syncthreads();

  float W1 = 0.f, W2 = 0.f;
  for (int b = 0; b < nbPrep; ++b) {
    W1 += prep_part[b*4 + 2];
    W2 += prep_part[b*4 + 3];
  }

  const int   i   = blockIdx.x * blockDim.x + tid;
  const float si  = s_s[i],  y1i = s_y1[i], y2i = s_y2[i];
  const float w1i = s_u1[i], w2i = s_u2[i];

  float mx1=0,my1=0,qxx1=0,qxy1=0,qyy1=0;
  float mx2=0,my2=0,qxx2=0,qxy2=0,qyy2=0;
  #pragma unroll 4
  for (int j = 0; j < N; ++j) {           // LDS broadcast reads (uniform j)
    float dx = fabsf(si  - s_s[j]);
    float d1 = fabsf(y1i - s_y1[j]);
    float d2 = fabsf(y2i - s_y2[j]);
    float u1 = s_u1[j], u2 = s_u2[j];
    float a1 = dx * u1, b1 = d1 * u1;
    mx1 += a1;  my1 += b1;
    qxx1 = fmaf(a1, dx, qxx1); qxy1 = fmaf(a1, d1, qxy1); qyy1 = fmaf(b1, d1, qyy1);
    float a2 = dx * u2, b2 = d2 * u2;
    mx2 += a2;  my2 += b2;
    qxx2 = fmaf(a2, dx, qxx2); qxy2 = fmaf(a2, d2, qxy2); qyy2 = fmaf(b2, d2, qyy2);
  }

  // Normalize (weights were unnormalized; fold 1/W here).
  float inv1 = 1.0f / fmaxf(W1, EPS_W), inv2 = 1.0f / fmaxf(W2, EPS_W);
  mx1 *= inv1; my1 *= inv1;  mx2 *= inv2; my2 *= inv2;
  float wi1 = w1i * inv1, wi2 = w2i * inv2;
  float contrib[16] = {
    wi1*mx1, wi1*my1, wi1*mx1*my1, wi1*mx1*mx1, wi1*my1*my1,
    wi1*qxx1*inv1, wi1*qxy1*inv1, wi1*qyy1*inv1,
    wi2*mx2, wi2*my2, wi2*mx2*my2, wi2*mx2*mx2, wi2*my2*my2,
    wi2*qxx2*inv2, wi2*qxy2*inv2, wi2*qyy2*inv2 };

  __shared__ float red[4][16];
  const int wave = tid >> 5, lane = tid & 31;
  #pragma unroll
  for (int q = 0; q < 16; ++q) {
    float s = wave_sum_wmma(contrib[q]);   // v_wmma_f32_16x16x4_f32
    if (lane == 0) red[wave][q] = s;
  }
  __syncthreads();
  if (tid == 0) {
    #pragma unroll
    for (int q = 0; q < 16; ++q) {
      float s = 0.f;
      for (int wv = 0; wv < 4; ++wv) s += red[wv][q];
      pair_part[blockIdx.x * 16 + q] = s;  // deterministic (no atomics)
    }
  }
}

// ---------------- Kernel 3: final scalar -----------------------------------
__global__ void k_final(const float* __restrict__ prep_part, int nbPrep,
                        const float* __restrict__ pair_part, int nbPair,
                        float* __restrict__ out) {
  float cw = 0.f, wsum = 0.f;
  for (int b = 0; b < nbPrep; ++b) {
    cw   += prep_part[b*4 + 0];
    wsum += prep_part[b*4 + 1];
  }
  float acc[16];
  for (int q = 0; q < 16; ++q) acc[q] = 0.f;
  for (int b = 0; b < nbPair; ++b)
    for (int q = 0; q < 16; ++q) acc[q] += pair_part[b*16 + q];

  float total = cw / fmaxf(wsum, EPS_W);
  for (int t = 0; t < 2; ++t) {
    const float* a = acc + t*8;
    float gx = a[0], gy = a[1], hxy = a[2], hxx = a[3], hyy = a[4];
    float sxx = a[5], sxy = a[6], syy = a[7];
    float dcov = sxy - 2.f*hxy + gx*gy;      // double-centering identity
    float dvx  = sxx - 2.f*hxx + gx*gx;
    float dvy  = syy - 2.f*hyy + gy*gy;
    bool  ok   = (dvx > EPS_VAR) && (dvy > EPS_VAR);
    float den  = sqrtf(ok ? dvx*dvy : 1.0f);
    float dcor = ok ? sqrtf(fabsf(dcov) / den) : 0.0f;
    total += DISCO_LAMBDA * dcor;
  }
  out[0] = total;
}

extern "C" void kernel_launch(void* const* d_in, const int* in_sizes, int n_in,
                              void* d_out, int out_size, void* d_ws, size_t ws_size,
                              hipStream_t stream) {
  (void)n_in; (void)out_size; (void)ws_size;
  const float* logits = (const float*)d_in[0];
  const int*   target = (const int*)  d_in[1];
  const float* weight = (const float*)d_in[2];
  const float* mass1  = (const float*)d_in[3];
  const float* mass2  = (const float*)d_in[4];
  const int N = in_sizes[2];                 // 4096

  float* wsf       = (float*)d_ws;
  const int nbPrep = N / 256;                // 16
  const int nbPair = N / 128;                // 32
  float* prep_part = wsf;                    // nbPrep*4 floats
  float* pair_part = wsf + nbPrep * 4;       // nbPair*16 floats
  float* scores    = wsf + 1024;             // 16B-aligned arrays
  float* w1a       = scores + N;
  float* w2a       = w1a + N;

  k_prep<<<nbPrep, 256, 0, stream>>>(logits, target, weight, mass1, mass2,
                                     scores, w1a, w2a, prep_part);
  size_t sh = (size_t)5 * N * sizeof(float); // 80KB dynamic LDS (WGP: 320KB)
  k_pairs<<<nbPair, 128, sh, stream>>>(scores, mass1, mass2, w1a, w2a,
                                       prep_part, nbPrep, N, pair_part);
  k_final<<<1, 1, 0, stream>>>(prep_part, nbPrep, pair_part, nbPair,
                               (float*)d_out);
}